// KGAT_transd_64106681860798
// MI455X (gfx1250) — compile-verified
//
#include <hip/hip_runtime.h>
#include <hip/hip_bf16.h>

typedef float v2f __attribute__((ext_vector_type(2)));
typedef float v8f __attribute__((ext_vector_type(8)));

#define EPS 1e-12f
#define NEG_SLOPE 0.01f

// ---------------------------------------------------------------------------
// Helpers
// ---------------------------------------------------------------------------
__device__ __forceinline__ float waveReduceSum(float v) {
#pragma unroll
  for (int m = 16; m >= 1; m >>= 1) v += __shfl_xor(v, m, 32);
  return v;
}

// Monotonic unsigned encoding of float for atomicMax over signed floats.
__device__ __forceinline__ unsigned encodeF32(float x) {
  int i = __float_as_int(x);
  return (i >= 0) ? ((unsigned)i | 0x80000000u) : ~(unsigned)i;
}
__device__ __forceinline__ float decodeF32(unsigned u) {
  return (u & 0x80000000u) ? __uint_as_float(u & 0x7FFFFFFFu)
                           : __uint_as_float(~u);
}

__device__ __forceinline__ float lrelu(float x) {
  return x >= 0.f ? x : NEG_SLOPE * x;
}

// ---------------------------------------------------------------------------
// Init kernels
// ---------------------------------------------------------------------------
__global__ void init_seg_kernel(unsigned* __restrict__ mEnc,
                                float* __restrict__ ssum, int n) {
  int i = blockIdx.x * blockDim.x + threadIdx.x;
  if (i < n) { mEnc[i] = 0u; ssum[i] = 0.f; }  // 0u decodes below any real att
}

__global__ void zero_f32_kernel(float* __restrict__ p, size_t n) {
  size_t i = (size_t)blockIdx.x * blockDim.x + threadIdx.x;
  if (i < n) p[i] = 0.f;
}

// Copy ent_embed (64 cols) into out columns [0,64) of the 112-wide output.
__global__ void copy_embed_kernel(const float* __restrict__ ent,
                                  float* __restrict__ out, int nNodes) {
  size_t i = (size_t)blockIdx.x * blockDim.x + threadIdx.x;
  if (i >= (size_t)nNodes * 64) return;
  size_t nrow = i >> 6;
  int c = (int)(i & 63);
  out[nrow * 112 + c] = ent[i];
}

// ---------------------------------------------------------------------------
// Attention: one wave32 per edge, 2 elements (float2) per lane over D=64.
//   Mrt = l2n(t_e + <t_e,t_p> r_p); Mrh = l2n(h_e + <h_e,h_p> r_p)
//   att = <Mrt, tanh(Mrh + r_e)>; also atomicMax segment-max per dst.
// ---------------------------------------------------------------------------
__global__ void att_kernel(const int* __restrict__ src,
                           const int* __restrict__ dst,
                           const int* __restrict__ etype,
                           const float* __restrict__ ent_embed,
                           const float* __restrict__ ent_transfer,
                           const float* __restrict__ rel_embed,
                           const float* __restrict__ rel_transfer,
                           float* __restrict__ att,
                           unsigned* __restrict__ mEnc, int nEdges) {
  int e = blockIdx.x * (blockDim.x >> 5) + (threadIdx.x >> 5);
  if (e >= nEdges) return;
  int lane = threadIdx.x & 31;

  int si = src[e], di = dst[e], ri = etype[e];
  float2 te = ((const float2*)(ent_embed    + (size_t)si * 64))[lane];
  float2 he = ((const float2*)(ent_embed    + (size_t)di * 64))[lane];
  float2 tp = ((const float2*)(ent_transfer + (size_t)si * 64))[lane];
  float2 hp = ((const float2*)(ent_transfer + (size_t)di * 64))[lane];
  float2 re = ((const float2*)(rel_embed    + (size_t)ri * 64))[lane];
  float2 rp = ((const float2*)(rel_transfer + (size_t)ri * 64))[lane];

  float dt = waveReduceSum(te.x * tp.x + te.y * tp.y);
  float dh = waveReduceSum(he.x * hp.x + he.y * hp.y);

  float mrtx = te.x + dt * rp.x, mrty = te.y + dt * rp.y;
  float mrhx = he.x + dh * rp.x, mrhy = he.y + dh * rp.y;

  float nt = waveReduceSum(mrtx * mrtx + mrty * mrty);
  float nh = waveReduceSum(mrhx * mrhx + mrhy * mrhy);
  float st = 1.f / fmaxf(sqrtf(nt), EPS);
  float sh = 1.f / fmaxf(sqrtf(nh), EPS);

  float thx = tanhf(mrhx * sh + re.x);
  float thy = tanhf(mrhy * sh + re.y);
  float a = waveReduceSum(mrtx * st * thx + mrty * st * thy);

  if (lane == 0) {
    att[e] = a;
    atomicMax(&mEnc[di], encodeF32(a));
  }
}

// ex = exp(att - max[dst]); att <- ex (in place); s[dst] += ex
__global__ void softmax_exp_kernel(const int* __restrict__ dst,
                                   const unsigned* __restrict__ mEnc,
                                   float* __restrict__ att,
                                   float* __restrict__ ssum, int nEdges) {
  int e = blockIdx.x * blockDim.x + threadIdx.x;
  if (e >= nEdges) return;
  int di = dst[e];
  float m = decodeF32(mEnc[di]);           // segment non-empty here -> finite
  float ex = expf(att[e] - m);
  att[e] = ex;
  atomicAdd(&ssum[di], ex);
}

__global__ void softmax_div_kernel(const int* __restrict__ dst,
                                   const float* __restrict__ ssum,
                                   float* __restrict__ att, int nEdges) {
  int e = blockIdx.x * blockDim.x + threadIdx.x;
  if (e >= nEdges) return;
  att[e] = att[e] / fmaxf(ssum[dst[e]], EPS);
}

// ---------------------------------------------------------------------------
// Message passing: Nh[dst] += att_e * node[src].  Wave per edge.
// ---------------------------------------------------------------------------
__global__ void msg_kernel(const int* __restrict__ src,
                           const int* __restrict__ dst,
                           const float* __restrict__ att,
                           const float* __restrict__ node,
                           float* __restrict__ Nh, int nEdges, int d) {
  int e = blockIdx.x * (blockDim.x >> 5) + (threadIdx.x >> 5);
  if (e >= nEdges) return;
  int lane = threadIdx.x & 31;
  float a = att[e];
  const float* ns = node + (size_t)src[e] * d;
  float* nd = Nh + (size_t)dst[e] * d;
  for (int k = lane; k < d; k += 32) atomicAdd(&nd[k], a * ns[k]);
}

// ---------------------------------------------------------------------------
// Bi-interaction layer via V_WMMA_F32_16X16X4_F32.
// One wave per 16-node tile. A = (node+Nh) and (node*Nh) 16x4 fragments,
// B = W 4x16 fragments, K-loop over DIN in steps of 4, NT = DOUT/16 N-tiles.
// ISA fragment layout (CDNA5 ISA):
//   A 16x4 f32:  lane l: M=l&15; VGPR0->K=2*(l>>4), VGPR1->K=2*(l>>4)+1
//   B 4x16 f32:  lane l: N=l&15; VGPR0->K=2*(l>>4), VGPR1->K=2*(l>>4)+1
//   D 16x16 f32: VGPR r: lanes 0-15 M=r, lanes 16-31 M=r+8; N=l&15
// ---------------------------------------------------------------------------
template <int DIN, int DOUT>
__global__ void bi_layer_wmma_kernel(const float* __restrict__ node,
                                     const float* __restrict__ Nh,
                                     const float* __restrict__ W1,
                                     const float* __restrict__ b1,
                                     const float* __restrict__ W2,
                                     const float* __restrict__ b2,
                                     float* __restrict__ outNode, int nTiles) {
  constexpr int NT = DOUT / 16;
  int tile = blockIdx.x * (blockDim.x >> 5) + (threadIdx.x >> 5);
  if (tile >= nTiles) return;  // uniform per wave (tiles aligned to 16 rows)
  int lane = threadIdx.x & 31;
  int m = lane & 15;
  int kh = (lane >> 4) << 1;  // 0 for lanes 0-15, 2 for lanes 16-31

  size_t rowBase = (size_t)(tile * 16 + m) * DIN;

  v8f zero = {};
  v8f acc1[NT], acc2[NT];
#pragma unroll
  for (int nt = 0; nt < NT; ++nt) { acc1[nt] = zero; acc2[nt] = zero; }

#pragma unroll
  for (int k0 = 0; k0 < DIN; k0 += 4) {
    int k = k0 + kh;
    float2 nd = *(const float2*)(node + rowBase + k);
    float2 nh = *(const float2*)(Nh + rowBase + k);
    v2f a1; a1.x = nd.x + nh.x; a1.y = nd.y + nh.y;   // (node + N_h)
    v2f a2; a2.x = nd.x * nh.x; a2.y = nd.y * nh.y;   // (node * N_h)
#pragma unroll
    for (int nt = 0; nt < NT; ++nt) {
      int n = nt * 16 + m;
      v2f bw1; bw1.x = W1[(size_t)k * DOUT + n]; bw1.y = W1[(size_t)(k + 1) * DOUT + n];
      v2f bw2; bw2.x = W2[(size_t)k * DOUT + n]; bw2.y = W2[(size_t)(k + 1) * DOUT + n];
      acc1[nt] = __builtin_amdgcn_wmma_f32_16x16x4_f32(
          false, a1, false, bw1, (short)0, acc1[nt], false, false);
      acc2[nt] = __builtin_amdgcn_wmma_f32_16x16x4_f32(
          false, a2, false, bw2, (short)0, acc2[nt], false, false);
    }
  }

  int rofs = (lane >> 4) * 8;  // lanes 16-31 hold rows M+8
#pragma unroll
  for (int nt = 0; nt < NT; ++nt) {
    int col = nt * 16 + m;
    float bias1 = b1[col], bias2 = b2[col];
#pragma unroll
    for (int r = 0; r < 8; ++r) {
      float v = lrelu(acc1[nt][r] + bias1) + lrelu(acc2[nt][r] + bias2);
      outNode[(size_t)(tile * 16 + rofs + r) * DOUT + col] = v;
    }
  }
}

// ---------------------------------------------------------------------------
// L2-normalize node rows and write to output columns [colOff, colOff+d).
// One wave per node.
// ---------------------------------------------------------------------------
__global__ void norm_out_kernel(const float* __restrict__ node,
                                float* __restrict__ out, int nNodes, int d,
                                int colOff) {
  int nrow = blockIdx.x * (blockDim.x >> 5) + (threadIdx.x >> 5);
  if (nrow >= nNodes) return;
  int lane = threadIdx.x & 31;
  float v = (lane < d) ? node[(size_t)nrow * d + lane] : 0.f;
  float ss = waveReduceSum(v * v);
  float sc = 1.f / fmaxf(sqrtf(ss), EPS);
  if (lane < d) out[(size_t)nrow * 112 + colOff + lane] = v * sc;
}

// ---------------------------------------------------------------------------
// Launch
// ---------------------------------------------------------------------------
extern "C" void kernel_launch(void* const* d_in, const int* in_sizes, int n_in,
                              void* d_out, int out_size, void* d_ws,
                              size_t ws_size, hipStream_t stream) {
  const int* src   = (const int*)d_in[0];
  const int* dst   = (const int*)d_in[1];
  const int* etype = (const int*)d_in[2];
  const float* ent_embed    = (const float*)d_in[3];
  const float* ent_transfer = (const float*)d_in[4];
  const float* rel_embed    = (const float*)d_in[5];
  const float* rel_transfer = (const float*)d_in[6];
  const float* W1_0 = (const float*)d_in[7];
  const float* b1_0 = (const float*)d_in[8];
  const float* W2_0 = (const float*)d_in[9];
  const float* b2_0 = (const float*)d_in[10];
  const float* W1_1 = (const float*)d_in[11];
  const float* b1_1 = (const float*)d_in[12];
  const float* W2_1 = (const float*)d_in[13];
  const float* b2_1 = (const float*)d_in[14];
  float* out = (float*)d_out;

  const int E = in_sizes[0];
  const int N = in_sizes[3] / 64;
  const int nTiles = N / 16;  // 100000 / 16 = 6250 exactly

  // Workspace partition (all 4-byte aligned)
  float* att       = (float*)d_ws;                 // [E]
  unsigned* mEnc   = (unsigned*)(att + E);         // [N]
  float* ssum      = (float*)(mEnc + N);           // [N]
  float* Nh        = ssum + N;                     // [N*64]
  float* node1     = Nh + (size_t)N * 64;          // [N*32]
  float* node2     = node1 + (size_t)N * 32;       // [N*16]

  const int T = 256;
  const int wavesPerBlock = T / 32;
  int edgeWaveBlocks = (E + wavesPerBlock - 1) / wavesPerBlock;
  int tileBlocks     = (nTiles + wavesPerBlock - 1) / wavesPerBlock;
  int nodeWaveBlocks = (N + wavesPerBlock - 1) / wavesPerBlock;

  // --- attention + edge softmax ---
  init_seg_kernel<<<(N + T - 1) / T, T, 0, stream>>>(mEnc, ssum, N);
  att_kernel<<<edgeWaveBlocks, T, 0, stream>>>(
      src, dst, etype, ent_embed, ent_transfer, rel_embed, rel_transfer, att,
      mEnc, E);
  softmax_exp_kernel<<<(E + T - 1) / T, T, 0, stream>>>(dst, mEnc, att, ssum, E);
  softmax_div_kernel<<<(E + T - 1) / T, T, 0, stream>>>(dst, ssum, att, E);

  // --- output cols [0,64): raw ent_embed ---
  copy_embed_kernel<<<(int)(((size_t)N * 64 + T - 1) / T), T, 0, stream>>>(
      ent_embed, out, N);

  // --- layer 0: d 64 -> 32 ---
  zero_f32_kernel<<<(int)(((size_t)N * 64 + T - 1) / T), T, 0, stream>>>(
      Nh, (size_t)N * 64);
  msg_kernel<<<edgeWaveBlocks, T, 0, stream>>>(src, dst, att, ent_embed, Nh, E, 64);
  bi_layer_wmma_kernel<64, 32><<<tileBlocks, T, 0, stream>>>(
      ent_embed, Nh, W1_0, b1_0, W2_0, b2_0, node1, nTiles);
  norm_out_kernel<<<nodeWaveBlocks, T, 0, stream>>>(node1, out, N, 32, 64);

  // --- layer 1: d 32 -> 16 ---
  zero_f32_kernel<<<(int)(((size_t)N * 32 + T - 1) / T), T, 0, stream>>>(
      Nh, (size_t)N * 32);
  msg_kernel<<<edgeWaveBlocks, T, 0, stream>>>(src, dst, att, node1, Nh, E, 32);
  bi_layer_wmma_kernel<32, 16><<<tileBlocks, T, 0, stream>>>(
      node1, Nh, W1_1, b1_1, W2_1, b2_1, node2, nTiles);
  norm_out_kernel<<<nodeWaveBlocks, T, 0, stream>>>(node2, out, N, 16, 96);
}